// ProjectLayer_21071109554713
// MI455X (gfx1250) — compile-verified
//
#include <hip/hip_runtime.h>

// Problem constants (from reference)
#define B_      8
#define NCAM    5
#define J_      15
#define H_      128
#define W_      240
#define NBINS   128000          // 80*80*20
#define SLICE   (H_ * W_)       // 30720 floats
#define SLICE_BYTES (SLICE * 4) // 122880 bytes (fits 2x in 320KB LDS)

// Tiling
#define BLOCK   256             // 8 waves (wave32)
#define BPT     25              // bins per thread
#define CHUNK   (BLOCK * BPT)   // 6400 bins per workgroup
#define NCHUNK  (NBINS / CHUNK) // 20, exact

typedef unsigned int u32x4 __attribute__((ext_vector_type(4)));
typedef int          i32x8 __attribute__((ext_vector_type(8)));
typedef int          i32x4 __attribute__((ext_vector_type(4)));

#if defined(__has_builtin)
#if __has_builtin(__builtin_amdgcn_tensor_load_to_lds) && __has_builtin(__builtin_amdgcn_s_wait_tensorcnt)
#define USE_TDM 1
#endif
#endif

// Issue a TDM copy of one contiguous 240x128 f32 heatmap slice (120KB) into LDS.
// D# layout per CDNA5 ISA ch.8: group0 = {ctl, lds_addr, global_addr lo/hi+type},
// group1 = dims/strides/tile, groups 2/3 unused (2D tensor).
// This toolchain (clang-23 / therock-10.0) exposes the 6-arg builtin form:
//   (u32x4 g0, i32x8 g1, i32x4 g2, i32x4 g3, i32x8 extra, i32 cpol)
__device__ __forceinline__ void tdm_issue_slice(const float* gptr, unsigned ldsOff) {
#ifdef USE_TDM
    unsigned long long ga = (unsigned long long)(uintptr_t)gptr;
    u32x4 g0;
    g0.x = 1u;                                                  // count=1, user descriptor
    g0.y = ldsOff;                                              // lds_addr (bytes)
    g0.z = (unsigned)(ga & 0xFFFFFFFFull);                      // global_addr[31:0]
    g0.w = (unsigned)((ga >> 32) & 0x01FFFFFFull) | (2u << 30); // addr[56:32] | type=2
    i32x8 g1;
    g1[0] = 0x00020000;                          // wg_mask=0, data_size=2 (4B), no flags
    g1[1] = (int)(((unsigned)W_)    << 16);      // tensor_dim0[15:0] @bits63:48
    g1[2] = (int)(((unsigned)H_)    << 16);      // tensor_dim1[15:0] @bits95:80
    g1[3] = (int)(((unsigned)W_)    << 16);      // tile_dim0 @bits127:112
    g1[4] = (int)(unsigned)H_;                   // tile_dim1 @bits143:128, tile_dim2=0
    g1[5] = (int)(unsigned)W_;                   // tensor_dim0_stride[31:0]
    g1[6] = (int)(((unsigned)SLICE) << 16);      // tensor_dim1_stride[15:0] @bits223:208
    g1[7] = 0;
    i32x4 gz4 = {0, 0, 0, 0};                    // groups 2/3 unused (<=2D)
    i32x8 gz8 = {0, 0, 0, 0, 0, 0, 0, 0};
    __builtin_amdgcn_tensor_load_to_lds(g0, g1, gz4, gz4, gz8, 0);
#else
    (void)gptr; (void)ldsOff;
#endif
}

__global__ void __launch_bounds__(BLOCK)
project_bilinear_tdm_kernel(const float* __restrict__ hm,     // [B,NCAM,J,H,W]
                            const float* __restrict__ grid,   // [NCAM,NBINS,2]
                            float* __restrict__ out) {        // [B,J,NBINS]
    extern __shared__ float smem[];                           // 2 x SLICE floats
    const int tid   = threadIdx.x;
    const int chunk = blockIdx.x;
    const int j     = blockIdx.y;
    const int b     = blockIdx.z;
    const int bin0  = chunk * CHUNK;

    // Prime the pipeline: DMA camera-0 slice into buffer 0.
    if (tid == 0) {
        const float* p0 = hm + (((size_t)b * NCAM + 0) * J_ + j) * (size_t)SLICE;
        tdm_issue_slice(p0, 0u);
    }

    float acc[BPT];
#pragma unroll
    for (int i = 0; i < BPT; ++i) acc[i] = 0.0f;

    for (int n = 0; n < NCAM; ++n) {
        if (tid == 0) {
            if (n + 1 < NCAM) {
                // Buffer (n+1)&1 was last read in iteration n-1; all waves passed
                // the end-of-iteration barrier, so it is free to overwrite.
                const float* pn = hm + (((size_t)b * NCAM + (n + 1)) * J_ + j) * (size_t)SLICE;
                tdm_issue_slice(pn, (unsigned)(((n + 1) & 1) * SLICE_BYTES));
#ifdef USE_TDM
                __builtin_amdgcn_s_wait_tensorcnt(1);  // slice n done (TDM in-order)
#endif
            } else {
#ifdef USE_TDM
                __builtin_amdgcn_s_wait_tensorcnt(0);
#endif
            }
        }
        __syncthreads();

#ifndef USE_TDM
        // Fallback path (toolchain without the TDM builtin): cooperative copy.
        {
            const float4* src = (const float4*)(hm + (((size_t)b * NCAM + n) * J_ + j) * (size_t)SLICE);
            float4* dst = (float4*)(smem + (n & 1) * SLICE);
            for (int k = tid; k < SLICE / 4; k += BLOCK) dst[k] = src[k];
            __syncthreads();
        }
#endif
        const float* buf = smem + (n & 1) * SLICE;
        const float2* gp = (const float2*)grid + (size_t)n * NBINS;

#pragma unroll
        for (int i = 0; i < BPT; ++i) {
            const int bin = bin0 + i * BLOCK + tid;          // coalesced
            const float2 g = gp[bin];
            // align_corners=True mapping
            const float ix = (g.x + 1.0f) * (0.5f * (W_ - 1));
            const float iy = (g.y + 1.0f) * (0.5f * (H_ - 1));
            const float x0f = floorf(ix);
            const float y0f = floorf(iy);
            const float wx1 = ix - x0f, wy1 = iy - y0f;
            const float wx0 = 1.0f - wx1, wy0 = 1.0f - wy1;
            const int x0 = (int)x0f, y0 = (int)y0f;
            const int x1 = x0 + 1,  y1 = y0 + 1;
            const bool vx0 = (unsigned)x0 < (unsigned)W_;
            const bool vx1 = (unsigned)x1 < (unsigned)W_;
            const bool vy0 = (unsigned)y0 < (unsigned)H_;
            const bool vy1 = (unsigned)y1 < (unsigned)H_;
            const int x0c = x0 < 0 ? 0 : (x0 > W_ - 1 ? W_ - 1 : x0);
            const int x1c = x1 < 0 ? 0 : (x1 > W_ - 1 ? W_ - 1 : x1);
            const int y0c = y0 < 0 ? 0 : (y0 > H_ - 1 ? H_ - 1 : y0);
            const int y1c = y1 < 0 ? 0 : (y1 > H_ - 1 ? H_ - 1 : y1);
            // zeros padding == zero the weight of invalid corners
            const float w00 = (vx0 & vy0) ? wy0 * wx0 : 0.0f;
            const float w01 = (vx1 & vy0) ? wy0 * wx1 : 0.0f;
            const float w10 = (vx0 & vy1) ? wy1 * wx0 : 0.0f;
            const float w11 = (vx1 & vy1) ? wy1 * wx1 : 0.0f;
            const int r0 = y0c * W_, r1 = y1c * W_;
            acc[i] += w00 * buf[r0 + x0c] + w01 * buf[r0 + x1c]
                    + w10 * buf[r1 + x0c] + w11 * buf[r1 + x1c];
        }
        __syncthreads();   // all waves done reading buf before it is overwritten
    }

    // mean over cameras, clamp to [0,1], coalesced store
    float* op = out + ((size_t)b * J_ + j) * NBINS + bin0;
#pragma unroll
    for (int i = 0; i < BPT; ++i) {
        float v = acc[i] * (1.0f / NCAM);
        v = fminf(fmaxf(v, 0.0f), 1.0f);
        op[i * BLOCK + tid] = v;
    }
}

extern "C" void kernel_launch(void* const* d_in, const int* in_sizes, int n_in,
                              void* d_out, int out_size, void* d_ws, size_t ws_size,
                              hipStream_t stream) {
    (void)in_sizes; (void)n_in; (void)d_ws; (void)ws_size; (void)out_size;
    const float* hm   = (const float*)d_in[0];   // [8,5,15,128,240] f32
    const float* grid = (const float*)d_in[1];   // [5,128000,2] f32
    float* out        = (float*)d_out;           // [8,15,128000] f32

    dim3 grd(NCHUNK, J_, B_);                    // 20 x 15 x 8 = 2400 workgroups
    dim3 blk(BLOCK);
    size_t shmem = 2 * (size_t)SLICE_BYTES;      // 245760 B double buffer (<320KB/WGP)
    project_bilinear_tdm_kernel<<<grd, blk, shmem, stream>>>(hm, grid, out);
}